// DynamicLinearModel_37056977830426
// MI455X (gfx1250) — compile-verified
//
#include <hip/hip_runtime.h>
#include <math.h>

#define T_TOTAL 2000000
#define XDIM    32
#define CHUNK   2048
#define NCHUNK  ((T_TOTAL + CHUNK - 1) / CHUNK)   // 977; tail chunk = 1152 rows (72 tiles)

#define BLOCK1  128                // pass-1 block: 4 waves
#define WAVES1  (BLOCK1 / 32)
#define LROW    36                 // padded LDS row stride (dwords): bank-conflict-free,
                                   // rows 144B apart -> 16B aligned for b128 async writes
#define TILEDW  (16 * LROW)        // 576 dwords per staged 16x32 tile

typedef float v2f __attribute__((ext_vector_type(2)));
typedef float v8f __attribute__((ext_vector_type(8)));
typedef int   v4i __attribute__((ext_vector_type(4)));

// ---- gfx1250 async global->LDS path (guarded; falls back to direct loads) ----
#if defined(__has_builtin)
#if __has_builtin(__builtin_amdgcn_global_load_async_to_lds_b128) && \
    __has_builtin(__builtin_amdgcn_s_wait_asynccnt)
#define USE_ASYNC_LDS 1
#endif
#endif
#ifndef USE_ASYNC_LDS
#define USE_ASYNC_LDS 0
#endif

#if USE_ASYNC_LDS
typedef __attribute__((address_space(1))) v4i* g_v4i_ptr;   // global
typedef __attribute__((address_space(3))) v4i* l_v4i_ptr;   // LDS
#define ASYNC_CP16(gsrc, ldst)                                                  \
    __builtin_amdgcn_global_load_async_to_lds_b128((g_v4i_ptr)(gsrc),           \
                                                   (l_v4i_ptr)(ldst), 0, 0)
#endif

__device__ __forceinline__ float sigmoidf_dev(float x) {
    return 1.0f / (1.0f + __expf(-x));
}

// ---------------------------------------------------------------------------
// Pass 1: per chunk of 2048 rows, compute base[t] = X.eta + Z.zeta, u[t] = Z.gamma
// via V_WMMA_F32_16X16X4_F32 tiles.  Tiles are staged into LDS with
// global_load_async_to_lds_b128 (double-buffered, s_wait_asynccnt); WMMA
// A-operands then come from conflict-free ds_load_b64.  Also produces the chunk
// scan aggregate  S[c] = sum_l a^(C-1-l) * u[cC + l]  (weighted LDS reduction).
// ---------------------------------------------------------------------------
__global__ __launch_bounds__(BLOCK1)
void dlm_pass1(const float* __restrict__ X, const float* __restrict__ Z,
               const float* __restrict__ G, const float* __restrict__ eta,
               const float* __restrict__ zeta, const float* __restrict__ gamma,
               float* __restrict__ ws_base, float* __restrict__ ws_u,
               float* __restrict__ ws_S)
{
    __shared__ float u_lds[CHUNK];
    __shared__ float red[BLOCK1];
#if USE_ASYNC_LDS
    __shared__ float stage[WAVES1 * 2 * 2 * TILEDW];   // [wave][buf][mat][16x36]
#endif

    const int tid  = threadIdx.x;
    const int wave = tid >> 5;
    const int lane = tid & 31;
    const int half = lane >> 4;   // 0: K pair {0,1}; 1: K pair {2,3}
    const int m    = lane & 15;   // row-in-tile for A, column N for B/C/D
    const int c    = blockIdx.x;
    const int rows = min(CHUNK, T_TOTAL - c * CHUNK);   // 2048 or 1152, both %16==0
    const int nTiles = rows >> 4;

    for (int i = tid; i < CHUNK; i += BLOCK1) u_lds[i] = 0.0f;
    __syncthreads();

    // B for the Z chain: columns 0-7 broadcast zeta, columns 8-15 broadcast gamma
    const float* bzsel = (m < 8) ? zeta : gamma;

#if USE_ASYNC_LDS
    float* wstage = stage + wave * (2 * 2 * TILEDW);
    const int qrow = lane >> 3;        // chunk row group within tile
    const int qcol = (lane & 7) * 4;   // dword column base of this lane's 16B chunk

    auto issue = [&](int tile, int buf) {
        const long grow = (long)c * CHUNK + ((long)tile << 4);
        const float* gx = X + grow * XDIM;
        const float* gz = Z + grow * XDIM;
        float* lb = wstage + buf * (2 * TILEDW);
        __builtin_prefetch(gx + 4 * WAVES1 * 16 * XDIM, 0, 1);  // L2 stream-ahead
        __builtin_prefetch(gz + 4 * WAVES1 * 16 * XDIM, 0, 1);
#pragma unroll
        for (int i = 0; i < 4; ++i) {
            const int r  = qrow + 4 * i;            // tile row of this 16B chunk
            const int go = (lane + 32 * i) * 4;     // dword offset in 16x32 tile
            ASYNC_CP16(gx + go, lb + r * LROW + qcol);
            ASYNC_CP16(gz + go, lb + TILEDW + r * LROW + qcol);
        }
    };

    int buf = 0;
    if (wave < nTiles) issue(wave, 0);   // prefill
    for (int tile = wave; tile < nTiles; tile += WAVES1) {
        const int  nxt  = tile + WAVES1;
        const bool more = (nxt < nTiles);
        if (more) issue(nxt, buf ^ 1);
        if (more) __builtin_amdgcn_s_wait_asynccnt(8);   // prev tile's 8 copies done
        else      __builtin_amdgcn_s_wait_asynccnt(0);

        const long grow = (long)c * CHUNK + ((long)tile << 4);
        const float* lx = wstage + buf * (2 * TILEDW) + m * LROW + half * 2;
        const float* lz = lx + TILEDW;

        v8f cx = {0.f,0.f,0.f,0.f,0.f,0.f,0.f,0.f};
        v8f cz = {0.f,0.f,0.f,0.f,0.f,0.f,0.f,0.f};
#pragma unroll
        for (int k0 = 0; k0 < 8; ++k0) {
            v2f ax = *(const v2f*)(lx + k0 * 4);            // ds_load_b64, conflict-free
            v2f az = *(const v2f*)(lz + k0 * 4);
            v2f be = *(const v2f*)(eta   + k0 * 4 + half * 2);
            v2f bz = *(const v2f*)(bzsel + k0 * 4 + half * 2);
            cx = __builtin_amdgcn_wmma_f32_16x16x4_f32(false, ax, false, be,
                                                       (short)0, cx, false, false);
            cz = __builtin_amdgcn_wmma_f32_16x16x4_f32(false, az, false, bz,
                                                       (short)0, cz, false, false);
        }

        if (m == 0) {   // lanes 0/16: X.eta (all N equal) + Z.zeta (N=0)
            const long r = grow + half * 8;
            float4 o0 = {cx[0]+cz[0], cx[1]+cz[1], cx[2]+cz[2], cx[3]+cz[3]};
            float4 o1 = {cx[4]+cz[4], cx[5]+cz[5], cx[6]+cz[6], cx[7]+cz[7]};
            *(float4*)(ws_base + r)     = o0;
            *(float4*)(ws_base + r + 4) = o1;
        }
        if (m == 8) {   // lanes 8/24: Z.gamma (N=8)
            const long r    = grow + half * 8;
            const int  lloc = (tile << 4) + half * 8;
#pragma unroll
            for (int j = 0; j < 8; ++j) {
                ws_u[r + j]     = cz[j];
                u_lds[lloc + j] = cz[j];
            }
        }
        buf ^= 1;
    }
#else
    // Fallback: direct per-lane global_load_b64 operand fetch (round-1 scheme)
    for (int tile = wave; tile < nTiles; tile += WAVES1) {
        const int  lrow0 = tile << 4;
        const long grow  = (long)c * CHUNK + lrow0;
        const float* Xr = X + (grow + m) * XDIM + half * 2;
        const float* Zr = Z + (grow + m) * XDIM + half * 2;
        __builtin_prefetch(Xr + WAVES1 * 16 * XDIM, 0, 1);
        __builtin_prefetch(Zr + WAVES1 * 16 * XDIM, 0, 1);

        v8f cx = {0.f,0.f,0.f,0.f,0.f,0.f,0.f,0.f};
        v8f cz = {0.f,0.f,0.f,0.f,0.f,0.f,0.f,0.f};
#pragma unroll
        for (int k0 = 0; k0 < 8; ++k0) {
            v2f ax = *(const v2f*)(Xr + k0 * 4);
            v2f az = *(const v2f*)(Zr + k0 * 4);
            v2f be = *(const v2f*)(eta   + k0 * 4 + half * 2);
            v2f bz = *(const v2f*)(bzsel + k0 * 4 + half * 2);
            cx = __builtin_amdgcn_wmma_f32_16x16x4_f32(false, ax, false, be,
                                                       (short)0, cx, false, false);
            cz = __builtin_amdgcn_wmma_f32_16x16x4_f32(false, az, false, bz,
                                                       (short)0, cz, false, false);
        }
        if (m == 0) {
            const long r = grow + half * 8;
            float4 o0 = {cx[0]+cz[0], cx[1]+cz[1], cx[2]+cz[2], cx[3]+cz[3]};
            float4 o1 = {cx[4]+cz[4], cx[5]+cz[5], cx[6]+cz[6], cx[7]+cz[7]};
            *(float4*)(ws_base + r)     = o0;
            *(float4*)(ws_base + r + 4) = o1;
        }
        if (m == 8) {
            const long r    = grow + half * 8;
            const int  lloc = lrow0 + half * 8;
#pragma unroll
            for (int j = 0; j < 8; ++j) {
                ws_u[r + j]     = cz[j];
                u_lds[lloc + j] = cz[j];
            }
        }
    }
#endif
    __syncthreads();

    // Weighted reduction: S = sum_l a^(CHUNK-1-l) * u[l]
    const float a   = sigmoidf_dev(G[0]);
    const float l2a = log2f(a);
    float       w   = exp2f((float)tid * l2a);
    const float stp = exp2f((float)BLOCK1 * l2a);   // underflow to 0 is numerically fine
    float partial = 0.0f;
#pragma unroll
    for (int mm = 0; mm < CHUNK / BLOCK1; ++mm) {
        partial += u_lds[CHUNK - 1 - tid - BLOCK1 * mm] * w;
        w *= stp;
    }
    red[tid] = partial;
    __syncthreads();
    for (int off = BLOCK1 / 2; off > 0; off >>= 1) {
        if (tid < off) red[tid] += red[tid + off];
        __syncthreads();
    }
    if (tid == 0) ws_S[c] = red[0];
}

// ---------------------------------------------------------------------------
// Pass 2: serial scan over 977 chunk aggregates: theta_in[c+1] = a^C * theta_in[c] + S[c]
// ---------------------------------------------------------------------------
__global__ void dlm_pass2(const float* __restrict__ G, const float* __restrict__ S,
                          float* __restrict__ thin)
{
    if (threadIdx.x != 0 || blockIdx.x != 0) return;
    const float a  = sigmoidf_dev(G[0]);
    const float aC = exp2f((float)CHUNK * log2f(a));   // ~0 for a<1, CHUNK=2048
    float carry = 0.0f;
    for (int i = 0; i < NCHUNK; ++i) {
        thin[i] = carry;
        carry   = aC * carry + S[i];
    }
}

// ---------------------------------------------------------------------------
// Pass 3: per-chunk parallel scan of theta[t] = a*theta[t-1] + u[t-1], seeded with
// theta_in[c]; out[t] = theta[t] + base[t]. Affine-pair Hillis-Steele over LDS.
// ---------------------------------------------------------------------------
__global__ __launch_bounds__(256)
void dlm_pass3(const float* __restrict__ G,
               const float* __restrict__ ws_base, const float* __restrict__ ws_u,
               const float* __restrict__ ws_thin, float* __restrict__ out)
{
    __shared__ float aggw[256];
    __shared__ float aggs[256];

    const int  tid  = threadIdx.x;
    const int  c    = blockIdx.x;
    const int  rows = min(CHUNK, T_TOTAL - c * CHUNK);
    const long g0   = (long)c * CHUNK;
    const int  l0   = tid * 8;

    const float a  = sigmoidf_dev(G[0]);
    const float a2 = a * a;
    const float a4 = a2 * a2;
    const float a8 = a4 * a4;

    float u8[8], b8[8];
    if (l0 + 8 <= rows) {
        const float4 u0 = *(const float4*)(ws_u + g0 + l0);
        const float4 u1 = *(const float4*)(ws_u + g0 + l0 + 4);
        const float4 b0 = *(const float4*)(ws_base + g0 + l0);
        const float4 b1 = *(const float4*)(ws_base + g0 + l0 + 4);
        u8[0]=u0.x; u8[1]=u0.y; u8[2]=u0.z; u8[3]=u0.w;
        u8[4]=u1.x; u8[5]=u1.y; u8[6]=u1.z; u8[7]=u1.w;
        b8[0]=b0.x; b8[1]=b0.y; b8[2]=b0.z; b8[3]=b0.w;
        b8[4]=b1.x; b8[5]=b1.y; b8[6]=b1.z; b8[7]=b1.w;
    } else {
#pragma unroll
        for (int j = 0; j < 8; ++j) {
            const bool ok = (l0 + j) < rows;
            u8[j] = ok ? ws_u[g0 + l0 + j]    : 0.0f;
            b8[j] = ok ? ws_base[g0 + l0 + j] : 0.0f;
        }
    }

    // thread-local affine segment T(x) = a^8 * x + s
    float s = 0.0f;
#pragma unroll
    for (int j = 0; j < 8; ++j) s = a * s + u8[j];
    float w = a8;

    aggw[tid] = w; aggs[tid] = s;
    __syncthreads();
    for (int off = 1; off < 256; off <<= 1) {
        float pw = 1.0f, ps = 0.0f;
        if (tid >= off) { pw = aggw[tid - off]; ps = aggs[tid - off]; }
        __syncthreads();
        if (tid >= off) {
            s = w * ps + s;     // compose: later o earlier
            w = w * pw;
            aggw[tid] = w; aggs[tid] = s;
        }
        __syncthreads();
    }

    const float th_in = ws_thin[c];
    float x = (tid == 0) ? th_in : (aggw[tid - 1] * th_in + aggs[tid - 1]);
#pragma unroll
    for (int j = 0; j < 8; ++j) {
        if (l0 + j < rows) out[g0 + l0 + j] = x + b8[j];
        x = a * x + u8[j];
    }
}

// ---------------------------------------------------------------------------
extern "C" void kernel_launch(void* const* d_in, const int* in_sizes, int n_in,
                              void* d_out, int out_size, void* d_ws, size_t ws_size,
                              hipStream_t stream)
{
    const float* X    = (const float*)d_in[0];
    const float* Z    = (const float*)d_in[1];
    const float* G    = (const float*)d_in[2];
    const float* eta  = (const float*)d_in[3];
    const float* zeta = (const float*)d_in[4];
    const float* gam  = (const float*)d_in[5];
    float* out = (float*)d_out;

    float* ws      = (float*)d_ws;
    float* ws_base = ws;                               // T floats
    float* ws_u    = ws + (size_t)T_TOTAL;             // T floats
    float* ws_S    = ws + 2 * (size_t)T_TOTAL;         // NCHUNK floats
    float* ws_thin = ws_S + NCHUNK;                    // NCHUNK floats

    dlm_pass1<<<NCHUNK, BLOCK1, 0, stream>>>(X, Z, G, eta, zeta, gam,
                                             ws_base, ws_u, ws_S);
    dlm_pass2<<<1, 32, 0, stream>>>(G, ws_S, ws_thin);
    dlm_pass3<<<NCHUNK, 256, 0, stream>>>(G, ws_base, ws_u, ws_thin, out);
}